// WaveEmbeddingLayer_55989193671046
// MI455X (gfx1250) — compile-verified
//
#include <hip/hip_runtime.h>
#include <math.h>

// WaveEmbeddingLayer: wave[n,c,e] = sin((2*bit[n,c]-1) * c * 2pi/E)  (same for all e)
//                     prob[n,c]   = |wave[n,c,0]| * E
// Pure store-bandwidth kernel: ~672 MB out @ 23.3 TB/s  =>  ~29 us floor.
// One wave32 per (n,c) row: 512 floats = 128 float4 = 4 b128 NT stores per lane.

#define CHUNK 40

typedef float f32x4 __attribute__((ext_vector_type(4)));  // true vector type for NT builtin

__global__ __launch_bounds__(256) void wave_embed_kernel(
    const float* __restrict__ binary,   // nbits floats in {0,1}
    float* __restrict__ out_wave,       // nbits * E floats
    float* __restrict__ out_prob,       // nbits floats
    int nbits, int embed, float scale)  // scale = 2*pi/E
{
    const int tid  = blockIdx.x * blockDim.x + threadIdx.x;
    const int row  = tid >> 5;          // one wave32 per (n,c) pair
    const int lane = tid & 31;
    if (row >= nbits) return;

    // c = position within chunk; bits.reshape(B*S, CHUNK) is a flat reinterpret,
    // so binary[row] is exactly bits[n][c] with c = row % CHUNK.
    const int   c   = row % CHUNK;
    const float amp = 2.0f * binary[row] - 1.0f;          // broadcast load per wave
    const float v   = __sinf(amp * (float)c * scale);     // v_sin_f32, |arg| < 0.48 rad

    f32x4 vv;
    vv.x = v; vv.y = v; vv.z = v; vv.w = v;

    f32x4* rowp = reinterpret_cast<f32x4*>(out_wave + (size_t)row * (size_t)embed);
    const int q = embed >> 2;                             // float4 slots per row (128)

    // Coalesced, non-temporal b128 stores: lanes 0..31 cover 512B per step.
    #pragma unroll 4
    for (int k = lane; k < q; k += 32) {
        __builtin_nontemporal_store(vv, rowp + k);
    }

    if (lane == 0) {
        __builtin_nontemporal_store(fabsf(v) * (float)embed, out_prob + row);
    }
}

extern "C" void kernel_launch(void* const* d_in, const int* in_sizes, int n_in,
                              void* d_out, int out_size, void* d_ws, size_t ws_size,
                              hipStream_t stream) {
    (void)n_in; (void)d_ws; (void)ws_size;

    const float* binary = (const float*)d_in[0];
    const int nbits = in_sizes[0];                 // B * S * CHUNK = 327680
    if (nbits <= 0) return;

    // out = [wave (nbits*E) , prob (nbits)]  =>  E = out_size/nbits - 1
    const int embed = (int)((long long)out_size / (long long)nbits) - 1;   // 512
    const float scale = 6.28318530717958647692f / (float)embed;            // 2*pi/E

    float* out_wave = (float*)d_out;
    float* out_prob = (float*)d_out + (size_t)nbits * (size_t)embed;

    // One wave32 per row: total threads = nbits * 32, 8 waves per 256-thread block.
    const long long threads = (long long)nbits * 32ll;
    const int block = 256;
    const int grid  = (int)((threads + block - 1) / block);

    wave_embed_kernel<<<grid, block, 0, stream>>>(binary, out_wave, out_prob,
                                                  nbits, embed, scale);
}